// GTN_23622320128484
// MI455X (gfx1250) — compile-verified
//
#include <hip/hip_runtime.h>
#include <math.h>
#include <stdint.h>

#define N 2048
#define E 5
#define C 2
#define W_IN 512
#define W_OUT 64
#define NUM_CLASS 4
#define NUM_TARGET 1024
#define NN ((size_t)N * (size_t)N)

typedef float v2f __attribute__((ext_vector_type(2)));
typedef float v4f __attribute__((ext_vector_type(4)));
typedef float v8f __attribute__((ext_vector_type(8)));
typedef int v4i_vs __attribute__((vector_size(16)));  // matches builtin param pointee

// ---------------------------------------------------------------------------
// CDNA5 async global->LDS copy path (ASYNCcnt-tracked DMA into LDS).
// Guarded so the file compiles even if the toolchain lacks the builtins.
// ---------------------------------------------------------------------------
#if defined(__AMDGCN__) && __has_builtin(__builtin_amdgcn_global_load_async_to_lds_b128) && \
    __has_builtin(__builtin_amdgcn_s_wait_asynccnt)
#define HAVE_ASYNC 1
#endif

__device__ __forceinline__ void cp16(const float* g, float* l) {
#ifdef HAVE_ASYNC
    // Param types per compiler diagnostic: (int4 AS1*, int4 AS3*, imm offset, imm cpol).
    // Flat LDS address: low 32 bits are the LDS byte offset (ISA aperture rule).
    __builtin_amdgcn_global_load_async_to_lds_b128(
        (__attribute__((address_space(1))) v4i_vs*)(uintptr_t)g,
        (__attribute__((address_space(3))) v4i_vs*)(uint32_t)(uintptr_t)l, 0, 0);
#else
    *(v4f*)l = *(const v4f*)g;
#endif
}

// ---------------------------------------------------------------------------
// Tiny: softmax of the three [C,E] GTConv weights -> sm[3][C][E]
// ---------------------------------------------------------------------------
__global__ void k_softmax(const float* w0a, const float* w0b, const float* w1,
                          float* sm) {
    if (threadIdx.x == 0) {
        const float* srcs[3] = {w0a, w0b, w1};
        for (int s = 0; s < 3; ++s) {
            for (int c = 0; c < C; ++c) {
                float mx = -1e30f;
                for (int e = 0; e < E; ++e) mx = fmaxf(mx, srcs[s][c * E + e]);
                float ex[E];
                float den = 0.f;
                for (int e = 0; e < E; ++e) {
                    ex[e] = expf(srcs[s][c * E + e] - mx);
                    den += ex[e];
                }
                for (int e = 0; e < E; ++e)
                    sm[s * C * E + c * E + e] = ex[e] / den;
            }
        }
    }
}

// ---------------------------------------------------------------------------
// Mix A[N,N,E] with two coefficient sets -> a[C,N,N], b[C,N,N] in one pass
// ---------------------------------------------------------------------------
__global__ void k_mix2(const float* __restrict__ A, const float* __restrict__ sm,
                       float* __restrict__ outa, float* __restrict__ outb) {
    size_t p = (size_t)blockIdx.x * blockDim.x + threadIdx.x;
    float v[E];
#pragma unroll
    for (int e = 0; e < E; ++e) v[e] = A[p * E + e];
#pragma unroll
    for (int c = 0; c < C; ++c) {
        float sa = 0.f, sb = 0.f;
#pragma unroll
        for (int e = 0; e < E; ++e) {
            sa += sm[c * E + e] * v[e];          // s0a
            sb += sm[C * E + c * E + e] * v[e];  // s0b
        }
        outa[(size_t)c * NN + p] = sa;
        outb[(size_t)c * NN + p] = sb;
    }
}

__global__ void k_mix1(const float* __restrict__ A, const float* __restrict__ smc,
                       float* __restrict__ outa) {
    size_t p = (size_t)blockIdx.x * blockDim.x + threadIdx.x;
    float v[E];
#pragma unroll
    for (int e = 0; e < E; ++e) v[e] = A[p * E + e];
#pragma unroll
    for (int c = 0; c < C; ++c) {
        float s = 0.f;
#pragma unroll
        for (int e = 0; e < E; ++e) s += smc[c * E + e] * v[e];
        outa[(size_t)c * NN + p] = s;
    }
}

// ---------------------------------------------------------------------------
// Batched f32 WMMA GEMM: C[c] = Aop[c] @ Bop[c], 2048^3 per channel (c=0,1).
// NORM=1: A-operand transformed at fragment-read time:
//         a(i,k) = (i==k ? 0 : A[i,k]) * rdeg[c][k]  (fused column-normalize)
// Block = 256 threads (8 waves), tile 128x128, BK=32, double-buffered LDS
// filled by GLOBAL_LOAD_ASYNC_TO_LDS_B128 (ASYNCcnt), wave tile 64x32 ->
// 4x2 f32 16x16 accumulators, V_WMMA_F32_16X16X4_F32 inner loop.
// ---------------------------------------------------------------------------
#define BM 128
#define BN 128
#define BK 32
#define APAD 4  // A row stride 36 floats: 16B-aligned rows, conflict-free frags

__device__ __forceinline__ void stage_tiles(const float* __restrict__ Ab,
                                            const float* __restrict__ Bb,
                                            int i0, int j0, int kb, int t,
                                            float (*Asb)[BK + APAD],
                                            float (*Bsb)[BN]) {
    // A tile 128x32: 4 x 16B per thread (8 float4 per row)
#pragma unroll
    for (int r = 0; r < 4; ++r) {
        int idx = t + r * 256;
        int row = idx >> 3;
        int c4 = (idx & 7) * 4;
        cp16(Ab + (size_t)(i0 + row) * N + kb + c4, &Asb[row][c4]);
    }
    // B tile 32x128: 4 x 16B per thread (32 float4 per row, fully coalesced)
#pragma unroll
    for (int r = 0; r < 4; ++r) {
        int idx = t + r * 256;
        int row = idx >> 5;
        int c4 = (idx & 31) * 4;
        cp16(Bb + (size_t)(kb + row) * N + j0 + c4, &Bsb[row][c4]);
    }
}

template <int NORM>
__global__ __launch_bounds__(256) void k_gemm(const float* __restrict__ Aop,
                                              const float* __restrict__ Bop,
                                              float* __restrict__ Cout,
                                              const float* __restrict__ rdeg) {
    __shared__ alignas(16) float As[2][BM][BK + APAD];  // 2 x 18 KB
    __shared__ alignas(16) float Bs[2][BK][BN];         // 2 x 16 KB

    const int ch = blockIdx.z;
    const float* Ab = Aop + (size_t)ch * NN;
    const float* Bb = Bop + (size_t)ch * NN;
    float* Cb = Cout + (size_t)ch * NN;
    const float* rd = NORM ? (rdeg + ch * N) : nullptr;

    const int i0 = blockIdx.y * BM;
    const int j0 = blockIdx.x * BN;
    const int t = threadIdx.x;
    const int lane = t & 31;
    const int wave = t >> 5;
    const int wm0 = (wave & 1) * 64;   // wave M offset in block tile
    const int wn0 = (wave >> 1) * 32;  // wave N offset in block tile
    const int l15 = lane & 15;
    const int hi = lane >> 4;          // half-wave selector (K pairs)

    v8f acc[4][2];
    const v8f vzero = {0.f, 0.f, 0.f, 0.f, 0.f, 0.f, 0.f, 0.f};
#pragma unroll
    for (int mi = 0; mi < 4; ++mi)
#pragma unroll
        for (int ni = 0; ni < 2; ++ni) acc[mi][ni] = vzero;

    stage_tiles(Ab, Bb, i0, j0, 0, t, As[0], Bs[0]);

    for (int kb = 0; kb < N; kb += BK) {
        const int cur = (kb >> 5) & 1;
        const bool hasNext = (kb + BK) < N;
        if (hasNext)
            stage_tiles(Ab, Bb, i0, j0, kb + BK, t, As[cur ^ 1], Bs[cur ^ 1]);
#ifdef HAVE_ASYNC
        // in-order completion: <=8 outstanding means current stage landed
        if (hasNext) __builtin_amdgcn_s_wait_asynccnt(8);
        else         __builtin_amdgcn_s_wait_asynccnt(0);
#endif
        __syncthreads();

        float (*Acur)[BK + APAD] = As[cur];
        float (*Bcur)[BN] = Bs[cur];

#pragma unroll
        for (int k0 = 0; k0 < BK; k0 += 4) {
            const int kg = kb + k0 + 2 * hi;  // global k of component .x
            v2f rv;
            if (NORM) rv = *(const v2f*)(rd + kg);
            // A fragment: lane<16 -> (K0,K1), lane>=16 -> (K2,K3), M = l15
            v2f aF[4];
#pragma unroll
            for (int mi = 0; mi < 4; ++mi) {
                v2f a = *(const v2f*)&Acur[wm0 + mi * 16 + l15][k0 + 2 * hi];
                if (NORM) {
                    a.x *= rv.x;
                    a.y *= rv.y;
                    int rowg = i0 + wm0 + mi * 16 + l15;
                    if (rowg == kg) a.x = 0.f;      // zero diagonal
                    if (rowg == kg + 1) a.y = 0.f;
                }
                aF[mi] = a;
            }
            // B fragment: VGPR0 -> rows K0/K2, VGPR1 -> rows K1/K3, N = l15
            v2f bF[2];
#pragma unroll
            for (int ni = 0; ni < 2; ++ni) {
                int colb = wn0 + ni * 16 + l15;
                v2f b;
                b.x = Bcur[k0 + 2 * hi + 0][colb];
                b.y = Bcur[k0 + 2 * hi + 1][colb];
                bF[ni] = b;
            }
#pragma unroll
            for (int mi = 0; mi < 4; ++mi)
#pragma unroll
                for (int ni = 0; ni < 2; ++ni)
                    acc[mi][ni] = __builtin_amdgcn_wmma_f32_16x16x4_f32(
                        false, aF[mi], false, bF[ni], (short)0, acc[mi][ni],
                        false, false);
        }
        __syncthreads();
    }

    // write out: VGPR r holds rows (r, r+8) for lanes (<16, >=16)
#pragma unroll
    for (int mi = 0; mi < 4; ++mi)
#pragma unroll
        for (int ni = 0; ni < 2; ++ni) {
            int colg = j0 + wn0 + ni * 16 + l15;
#pragma unroll
            for (int r = 0; r < 8; ++r) {
                int rowg = i0 + wm0 + mi * 16 + hi * 8 + r;
                Cb[(size_t)rowg * N + colg] = acc[mi][ni][r];
            }
        }
}

// ---------------------------------------------------------------------------
// Reciprocal of off-diagonal column sums (optionally + self-loop), clamped.
// ---------------------------------------------------------------------------
__global__ void k_coldeg(const float* __restrict__ H, float* __restrict__ rdeg,
                         float addSelf) {
    int c = blockIdx.y;
    int j = blockIdx.x * blockDim.x + threadIdx.x;
    const float* Hb = H + (size_t)c * NN;
    float s = 0.f;
    for (int i = 0; i < N; ++i) s += Hb[(size_t)i * N + j];  // coalesced rows
    s -= Hb[(size_t)j * N + j];  // exclude diagonal
    s += addSelf;
    if (s == 0.f) s = 1e9f;
    rdeg[c * N + j] = 1.0f / s;
}

// ---------------------------------------------------------------------------
// Xw = X @ weight   [2048,512]@[512,64]
// ---------------------------------------------------------------------------
__global__ void k_xw(const float* __restrict__ X, const float* __restrict__ Wt,
                     float* __restrict__ Xw) {
    int n = blockIdx.x;
    int k = threadIdx.x;  // 64 threads
    float s = 0.f;
    for (int i = 0; i < W_IN; ++i)
        s += X[(size_t)n * W_IN + i] * Wt[i * W_OUT + k];
    Xw[n * W_OUT + k] = s;
}

// ---------------------------------------------------------------------------
// Xcat[m, c*64+k] = relu( rdeg2[c,m] * (sum_{n!=m} H2[c,n,m]*Xw[n,k] + Xw[m,k]) )
// ---------------------------------------------------------------------------
__global__ void k_gcn(const float* __restrict__ H2, const float* __restrict__ Xw,
                      const float* __restrict__ rdeg2, float* __restrict__ Xcat) {
    int tg = blockIdx.x * blockDim.x + threadIdx.x;  // C*N*64
    int k = tg & 63;
    int m = (tg >> 6) & (N - 1);
    int c = tg >> 17;
    const float* Hb = H2 + (size_t)c * NN;
    float s = 0.f;
    for (int n = 0; n < N; ++n) {
        float h = (n == m) ? 1.0f : Hb[(size_t)n * N + m];  // diag -> self-loop
        s += h * Xw[n * W_OUT + k];
    }
    s *= rdeg2[c * N + m];
    Xcat[m * (C * W_OUT) + c * W_OUT + k] = fmaxf(s, 0.f);
}

// ---------------------------------------------------------------------------
// X1 = relu(Xcat @ lin1_w^T + lin1_b)   [2048,128]@[128,64]
// ---------------------------------------------------------------------------
__global__ void k_lin1(const float* __restrict__ Xcat, const float* __restrict__ w,
                       const float* __restrict__ b, float* __restrict__ X1) {
    int tg = blockIdx.x * blockDim.x + threadIdx.x;  // N*64
    int o = tg & 63;
    int n = tg >> 6;
    float s = b[o];
    for (int j = 0; j < C * W_OUT; ++j)
        s += Xcat[n * (C * W_OUT) + j] * w[o * (C * W_OUT) + j];
    X1[n * W_OUT + o] = fmaxf(s, 0.f);
}

// ---------------------------------------------------------------------------
// y = X1[target_x] @ lin2_w^T + lin2_b ; per-target NLL of log_softmax
// ---------------------------------------------------------------------------
__global__ void k_head(const float* __restrict__ X1, const float* __restrict__ w2,
                       const float* __restrict__ b2, const int* __restrict__ tx,
                       const int* __restrict__ tgt, float* __restrict__ yout,
                       float* __restrict__ nll) {
    int t = blockIdx.x * blockDim.x + threadIdx.x;
    if (t >= NUM_TARGET) return;
    int row = tx[t];
    float y[NUM_CLASS];
#pragma unroll
    for (int cl = 0; cl < NUM_CLASS; ++cl) {
        float s = b2[cl];
        for (int j = 0; j < W_OUT; ++j)
            s += X1[row * W_OUT + j] * w2[cl * W_OUT + j];
        y[cl] = s;
        yout[t * NUM_CLASS + cl] = s;
    }
    float mx = y[0];
#pragma unroll
    for (int cl = 1; cl < NUM_CLASS; ++cl) mx = fmaxf(mx, y[cl]);
    float den = 0.f;
#pragma unroll
    for (int cl = 0; cl < NUM_CLASS; ++cl) den += expf(y[cl] - mx);
    float lse = mx + logf(den);
    nll[t] = lse - y[tgt[t]];
}

// Deterministic tree-reduced mean loss
__global__ void k_loss(const float* __restrict__ nll, float* __restrict__ out0) {
    __shared__ float sh[256];
    float s = 0.f;
    for (int i = threadIdx.x; i < NUM_TARGET; i += 256) s += nll[i];
    sh[threadIdx.x] = s;
    __syncthreads();
    for (int off = 128; off > 0; off >>= 1) {
        if ((int)threadIdx.x < off) sh[threadIdx.x] += sh[threadIdx.x + off];
        __syncthreads();
    }
    if (threadIdx.x == 0) out0[0] = sh[0] / (float)NUM_TARGET;
}

// ---------------------------------------------------------------------------
extern "C" void kernel_launch(void* const* d_in, const int* in_sizes, int n_in,
                              void* d_out, int out_size, void* d_ws, size_t ws_size,
                              hipStream_t stream) {
    const float* A      = (const float*)d_in[0];
    const float* X      = (const float*)d_in[1];
    const float* gtw0a  = (const float*)d_in[2];
    const float* gtw0b  = (const float*)d_in[3];
    const float* gtw1   = (const float*)d_in[4];
    const float* weight = (const float*)d_in[5];
    const float* lin1_w = (const float*)d_in[6];
    const float* lin1_b = (const float*)d_in[7];
    const float* lin2_w = (const float*)d_in[8];
    const float* lin2_b = (const float*)d_in[9];
    const int* target_x = (const int*)d_in[10];
    const int* target   = (const int*)d_in[11];
    float* out = (float*)d_out;  // [0]=loss, [1..4096]=y

    float* ws    = (float*)d_ws;
    float* buf0  = ws;                       // [C,N,N]  a, later a1
    float* buf1  = ws + (size_t)C * NN;      // [C,N,N]  b, later H2
    float* buf2  = ws + (size_t)2 * C * NN;  // [C,N,N]  H
    float* sm    = ws + (size_t)3 * C * NN;  // 30 softmax coeffs
    float* rdeg1 = sm + 32;                  // [C,N]
    float* rdeg2 = rdeg1 + C * N;            // [C,N]
    float* Xw    = rdeg2 + C * N;            // [N,64]
    float* Xcat  = Xw + N * W_OUT;           // [N,128]
    float* X1    = Xcat + N * C * W_OUT;     // [N,64]
    float* nll   = X1 + N * W_OUT;           // [1024]

    dim3 gGemm(N / BN, N / BM, C);

    k_softmax<<<1, 32, 0, stream>>>(gtw0a, gtw0b, gtw1, sm);
    k_mix2<<<(N * N) / 256, 256, 0, stream>>>(A, sm, buf0, buf1);        // a, b
    k_gemm<0><<<gGemm, 256, 0, stream>>>(buf0, buf1, buf2, nullptr);     // H = a@b
    k_mix1<<<(N * N) / 256, 256, 0, stream>>>(A, sm + 2 * C * E, buf0);  // a1
    k_coldeg<<<dim3(N / 256, C), 256, 0, stream>>>(buf2, rdeg1, 0.0f);
    k_gemm<1><<<gGemm, 256, 0, stream>>>(buf2, buf0, buf1, rdeg1);       // H2 = Hn@a1
    k_coldeg<<<dim3(N / 256, C), 256, 0, stream>>>(buf1, rdeg2, 1.0f);
    k_xw<<<N, W_OUT, 0, stream>>>(X, weight, Xw);
    k_gcn<<<(C * N * W_OUT) / 256, 256, 0, stream>>>(buf1, Xw, rdeg2, Xcat);
    k_lin1<<<(N * W_OUT) / 256, 256, 0, stream>>>(Xcat, lin1_w, lin1_b, X1);
    k_head<<<NUM_TARGET / 256, 256, 0, stream>>>(X1, lin2_w, lin2_b, target_x,
                                                 target, out + 1, nll);
    k_loss<<<1, 256, 0, stream>>>(nll, out);
}